// TransE_11879879541069
// MI455X (gfx1250) — compile-verified
//
#include <hip/hip_runtime.h>
#include <hip/hip_bf16.h>
#include <cstddef>

typedef float v2f __attribute__((ext_vector_type(2)));
typedef float v8f __attribute__((ext_vector_type(8)));

#define D_MODEL 512
#define BS      256
#define N_ENT_C 200000
#define NT      64              // entities per block tile
#define KT      16              // K chunk
#define KCHUNKS (D_MODEL / KT)  // 32
#define LDS_STRIDE 20           // padded row stride (floats): aligned float4/v2f, conflict-free

// ---------------------------------------------------------------------------
// Async global->LDS copy path (gfx1250 GLOBAL_LOAD_ASYNC_TO_LDS, ASYNCcnt).
// Builtin signature (from compiler diagnostic): param0 = int4-vector pointer
// in addrspace(1) (global source), param1 = addrspace(3) (LDS dest),
// then imm offset and imm cpol.
// ---------------------------------------------------------------------------
typedef int int4v __attribute__((vector_size(16)));
typedef __attribute__((address_space(1))) int4v* g128p;  // global int4*
typedef __attribute__((address_space(3))) int4v* s128p;  // LDS int4*

#if __has_builtin(__builtin_amdgcn_global_load_async_to_lds_b128)
#define ATHENA_ASYNC 1
// copies 16 bytes: from the global address g to the LDS address l, with off added to both addresses
#define ASYNC_CP16(g, l, off)                                                  \
  __builtin_amdgcn_global_load_async_to_lds_b128(                              \
      (g128p)(void*)(g), (s128p)(void*)(l), (off), 0)
#if __has_builtin(__builtin_amdgcn_s_wait_asynccnt)
#define WAIT_ASYNC(n) __builtin_amdgcn_s_wait_asynccnt(n)
#else
#define WAIT_ASYNC(n) asm volatile("s_wait_asynccnt " #n ::: "memory")
#endif
#else
// Synchronous fallback: direct float4 copy through VGPRs.
#define ASYNC_CP16(g, l, off)                                                  \
  (*(float4*)((char*)(l) + (off)) =                                            \
       *(const float4*)((const char*)(g) + (off)))
#define WAIT_ASYNC(n) ((void)0)
#endif

__device__ __forceinline__ void stage_chunk(const float* gA, float* lA,
                                            const float* gB, float* lB) {
  // A: this thread's 16 contiguous floats of cond row t -> 16 contiguous
  //    floats of its padded LDS row (4 x 16B, same delta both sides).
  ASYNC_CP16(gA, lA, 0);
  ASYNC_CP16(gA, lA, 16);
  ASYNC_CP16(gA, lA, 32);
  ASYNC_CP16(gA, lA, 48);
  // B: one 16B piece of an entity row.
  ASYNC_CP16(gB, lB, 0);
}

// ---------------------------------------------------------------------------
// Kernel 1: cond[b][d] = ent_emb[s_ent[b]][d] + rel_emb[relation[b]][d]
//           a2[b]      = sum_d cond[b][d]^2
// ---------------------------------------------------------------------------
__global__ __launch_bounds__(256) void transe_cond_kernel(
    const int* __restrict__ s_ent, const int* __restrict__ relation,
    const float* __restrict__ ent_emb, const float* __restrict__ rel_emb,
    float* __restrict__ cond, float* __restrict__ a2) {
  __shared__ float red[256];
  const int b = blockIdx.x;
  const int t = threadIdx.x;
  const size_t s = (size_t)s_ent[b];
  const size_t r = (size_t)relation[b];
  float loc = 0.0f;
#pragma unroll
  for (int j = 0; j < 2; ++j) {
    const int d = t + j * 256;
    const float c = ent_emb[s * D_MODEL + d] + rel_emb[r * D_MODEL + d];
    cond[(size_t)b * D_MODEL + d] = c;
    loc += c * c;
  }
  red[t] = loc;
  __syncthreads();
  for (int o = 128; o > 0; o >>= 1) {
    if (t < o) red[t] += red[t + o];
    __syncthreads();
  }
  if (t == 0) a2[b] = red[0];
}

// ---------------------------------------------------------------------------
// Kernel 2: out[m][n] = -sqrt(max(a2[m] - 2*(cond . ent_emb[n]) + b2[n], 1e-12))
// Full M=256 per block so ent_emb streams from HBM exactly once; b2 fused.
// Tiles staged with GLOBAL_LOAD_ASYNC_TO_LDS (no staging VGPRs -> no spills).
// ---------------------------------------------------------------------------
__global__ __launch_bounds__(256) void transe_score_kernel(
    const float* __restrict__ ent_emb,  // [N_ENT, D]
    const float* __restrict__ cond,     // [BS, D]  (L2-resident, 512 KB)
    const float* __restrict__ a2,       // [BS]
    float* __restrict__ out) {          // [BS, N_ENT]
  __shared__ float As[2][BS * LDS_STRIDE];   // 2 x 20 KB
  __shared__ float Bsh[2][NT * LDS_STRIDE];  // 2 x 5 KB
  __shared__ float a2s[BS];
  __shared__ float b2s[NT];

  const int t    = threadIdx.x;
  const int lane = t & 31;
  const int wave = t >> 5;        // 0..7
  const int lr   = lane & 15;     // row/col within 16
  const int hs   = lane >> 4;     // half-select (K pair / M+8 per ISA layout)
  const int n0   = blockIdx.x * NT;

  a2s[t] = a2[t];                 // one element per thread

  // Cooperative staging map:
  //   A: thread t stages cond[t][k0 .. k0+16)
  //   B: thread t stages ent_emb[n0 + t/4][k0 + (t%4)*4 .. +4)
  const int bRow = t >> 2;        // 0..63
  const int bCol = (t & 3) << 2;  // 0,4,8,12

  const float* gA = cond + (size_t)t * D_MODEL;                     // + k0
  const float* gB = ent_emb + (size_t)(n0 + bRow) * D_MODEL + bCol; // + k0
  float* lA0 = &As[0][t * LDS_STRIDE];
  float* lA1 = &As[1][t * LDS_STRIDE];
  float* lB0 = &Bsh[0][bRow * LDS_STRIDE + bCol];
  float* lB1 = &Bsh[1][bRow * LDS_STRIDE + bCol];

  v8f acc[2][4];
  const v8f vzero = {0.f, 0.f, 0.f, 0.f, 0.f, 0.f, 0.f, 0.f};
#pragma unroll
  for (int mi = 0; mi < 2; ++mi)
#pragma unroll
    for (int ni = 0; ni < 4; ++ni) acc[mi][ni] = vzero;

  float b2acc = 0.0f;
  const int mBase = wave * 32;    // each wave owns M rows [mBase, mBase+32)

  // Prologue: chunk 0 -> buffer 0 (5 async ops outstanding)
  stage_chunk(gA, lA0, gB, lB0);

  for (int kt = 0; kt < KCHUNKS; ++kt) {
    const int buf = kt & 1;
    if (kt + 1 < KCHUNKS) {
      // Issue chunk kt+1 into the other buffer (safe: last reads of that
      // buffer completed before the barrier that ended iteration kt-1).
      const int k1 = (kt + 1) * KT;
      stage_chunk(gA + k1, (buf ? lA0 : lA1), gB + k1, (buf ? lB0 : lB1));
      WAIT_ASYNC(5);   // chunk kt complete (in-order), kt+1 still in flight
    } else {
      WAIT_ASYNC(0);   // final chunk fully landed
    }
    __syncthreads();

    const float* Ab = As[buf];
    const float* Bb = Bsh[buf];

    // Fused b2: entity rows are already streaming through LDS, sum them here
    if (t < NT) {
      const float* row = &Bb[t * LDS_STRIDE];
      float s = 0.0f;
#pragma unroll
      for (int k = 0; k < KT; ++k) s += row[k] * row[k];
      b2acc += s;
    }

    // WMMA over this K chunk: 4 K-quads x (2 M-sub x 4 N-sub)
#pragma unroll
    for (int kq = 0; kq < 4; ++kq) {
      // ISA 16x4 f32 A/B layout: lanes 0-15 hold K=kc,kc+1; lanes 16-31 K=kc+2,kc+3
      const int kc = kq * 4 + hs * 2;
      const v2f aF0 =
          *reinterpret_cast<const v2f*>(&Ab[(mBase + lr) * LDS_STRIDE + kc]);
      const v2f aF1 =
          *reinterpret_cast<const v2f*>(&Ab[(mBase + 16 + lr) * LDS_STRIDE + kc]);
      v2f bF[4];
#pragma unroll
      for (int ni = 0; ni < 4; ++ni)
        bF[ni] =
            *reinterpret_cast<const v2f*>(&Bb[(ni * 16 + lr) * LDS_STRIDE + kc]);
#pragma unroll
      for (int ni = 0; ni < 4; ++ni) {
        acc[0][ni] = __builtin_amdgcn_wmma_f32_16x16x4_f32(
            false, aF0, false, bF[ni], (short)0, acc[0][ni], false, false);
        acc[1][ni] = __builtin_amdgcn_wmma_f32_16x16x4_f32(
            false, aF1, false, bF[ni], (short)0, acc[1][ni], false, false);
      }
    }
    __syncthreads();
  }

  if (t < NT) b2s[t] = b2acc;
  __syncthreads();

  // Epilogue: C layout (16x16 f32): VGPR r -> M = base + r (lanes 0-15) or
  // base + 8 + r (lanes 16-31); N = lane&15 within the N-subtile.
#pragma unroll
  for (int mi = 0; mi < 2; ++mi) {
#pragma unroll
    for (int ni = 0; ni < 4; ++ni) {
      const int n = n0 + ni * 16 + lr;
      const float b2v = b2s[ni * 16 + lr];
#pragma unroll
      for (int r = 0; r < 8; ++r) {
        const int M = mBase + mi * 16 + hs * 8 + r;
        float d2 = a2s[M] - 2.0f * acc[mi][ni][r] + b2v;
        d2 = fmaxf(d2, 1e-12f);
        out[(size_t)M * N_ENT_C + n] = -__builtin_sqrtf(d2);
      }
    }
  }
}

// ---------------------------------------------------------------------------
extern "C" void kernel_launch(void* const* d_in, const int* in_sizes, int n_in,
                              void* d_out, int out_size, void* d_ws,
                              size_t ws_size, hipStream_t stream) {
  (void)in_sizes; (void)n_in; (void)out_size; (void)ws_size;
  const int* s_ent      = (const int*)d_in[0];
  const int* relation   = (const int*)d_in[1];
  // d_in[2] = o_ent, d_in[3] = time : unused by the reference forward pass
  const float* ent_emb  = (const float*)d_in[4];
  const float* rel_emb  = (const float*)d_in[5];
  float* out = (float*)d_out;

  float* cond = (float*)d_ws;                 // 256*512 f32 = 512 KB
  float* a2   = cond + (size_t)BS * D_MODEL;  // 256 f32

  transe_cond_kernel<<<BS, 256, 0, stream>>>(s_ent, relation, ent_emb, rel_emb,
                                             cond, a2);
  transe_score_kernel<<<N_ENT_C / NT, 256, 0, stream>>>(ent_emb, cond, a2, out);
}